// CCN_63299228009054
// MI455X (gfx1250) — compile-verified
//
#include <hip/hip_runtime.h>
#include <stdint.h>

#define B_ 64
#define N_ 1024
#define D_ 128
#define K_ 6
#define F_ 768            // K_ * D_
#define KC 32             // K-chunk in LDS (32 | 128 => neighbor slot constant per chunk)
#define NCHUNK (F_ / KC)  // 24
#define MROWS 128         // output rows per block
#define LSTR (KC + 4)     // LDS row stride (36 words = 144B, 16B aligned, conflict-free)

typedef __attribute__((ext_vector_type(2))) float v2f;
typedef __attribute__((ext_vector_type(8))) float v8f;

// ---- CDNA5 async global->LDS DMA (ASYNCcnt-tracked), per-lane 16B transfer ----
__device__ __forceinline__ void async_b128(uint32_t lds_addr, const float* g) {
    asm volatile("global_load_async_to_lds_b128 %0, %1, off"
                 :: "v"(lds_addr), "v"(g) : "memory");
}
__device__ __forceinline__ void wait_async0() {
    asm volatile("s_wait_asynccnt 0x0" ::: "memory");
}

// ---------------- kernel 1: F2_0 = loc[0] @ W2d^T + b2d  (1024 x 128) ----------------
__global__ __launch_bounds__(256) void f2_kernel(const float* __restrict__ loc,
                                                 const float* __restrict__ W2d,
                                                 const float* __restrict__ b2d,
                                                 float* __restrict__ F2_0) {
    int t = blockIdx.x * 256 + threadIdx.x;
    if (t >= N_ * D_) return;
    int n = t >> 7, d = t & 127;
    float x = loc[n * 2 + 0];
    float y = loc[n * 2 + 1];
    F2_0[t] = fmaf(x, W2d[d * 2 + 0], fmaf(y, W2d[d * 2 + 1], b2d[d]));
}

// ---------------- kernel 2: per-batch 6-NN over 2-D points (4 blocks / batch) ----------------
__global__ __launch_bounds__(256) void knn_kernel(const float* __restrict__ loc,
                                                  int* __restrict__ nbr) {
    __shared__ float2 pts[N_];
    const int b  = blockIdx.x >> 2;
    const int i  = ((blockIdx.x & 3) << 8) + threadIdx.x;   // query index
    const float* lb = loc + (size_t)b * N_ * 2;
    for (int j = threadIdx.x; j < N_; j += 256)
        pts[j] = make_float2(lb[2 * j], lb[2 * j + 1]);
    __syncthreads();

    const float qx = pts[i].x, qy = pts[i].y;
    float bd[K_];
    int   bi[K_];
#pragma unroll
    for (int m = 0; m < K_; ++m) { bd[m] = 3.4e38f; bi[m] = 0; }
    for (int j = 0; j < N_; ++j) {
        float dx = pts[j].x - qx;
        float dy = pts[j].y - qy;
        float d2 = fmaf(dx, dx, dy * dy);
        if (d2 < bd[K_ - 1]) {            // strict <  => equal keys keep lower index
            bd[K_ - 1] = d2; bi[K_ - 1] = j;
#pragma unroll
            for (int s = K_ - 1; s > 0; --s) {
                if (bd[s] < bd[s - 1]) {
                    float td = bd[s]; bd[s] = bd[s - 1]; bd[s - 1] = td;
                    int   ti = bi[s]; bi[s] = bi[s - 1]; bi[s - 1] = ti;
                }
            }
        }
    }
    int* o = nbr + ((size_t)b * N_ + i) * K_;
#pragma unroll
    for (int m = 0; m < K_; ++m) o[m] = bi[m];
}

// ---------------- kernel 3: depot row  h[b,0,:] ----------------
__global__ __launch_bounds__(256) void dep_kernel(const float* __restrict__ depot,
                                                  const float* __restrict__ Wdep,
                                                  const float* __restrict__ bdep,
                                                  float* __restrict__ h) {
    int t = blockIdx.x * 256 + threadIdx.x;
    if (t >= B_ * D_) return;
    int b = t >> 7, d = t & 127;
    float v = fmaf(depot[b * 2 + 0], Wdep[d * 2 + 0],
              fmaf(depot[b * 2 + 1], Wdep[d * 2 + 1], bdep[d]));
    v = v > 0.0f ? v : 0.01f * v;
    h[((size_t)b * (N_ + 1)) * D_ + d] = v;
}

// ---------------- kernel 4: fused gather + (65536x768)@(768x128) WMMA GEMM ----------------
// block: 256 threads = 8 waves; 128 rows x 128 cols; wave w owns cols [16w,16w+16)
// and all 8 M-tiles (8 independent accumulators). Double-buffered async->LDS staging.
__global__ __launch_bounds__(256) void gemm_kernel(
    const float* __restrict__ F2_0, const int* __restrict__ nbr,
    const float* __restrict__ Wnb,  const float* __restrict__ bnb,
    const float* __restrict__ loc,  const float* __restrict__ deadline,
    const float* __restrict__ W3d,  const float* __restrict__ b3d,
    float* __restrict__ h) {

    __shared__ __align__(16) float As[2][MROWS][LSTR];
    __shared__ __align__(16) float Bs[2][D_][LSTR];
    __shared__ int nb_s[MROWS * K_];

    const int tid = threadIdx.x;
    const int r0  = blockIdx.x * MROWS;        // 128 | 1024 => single batch per block

    for (int t = tid; t < MROWS * K_; t += 256)
        nb_s[t] = nbr[(size_t)r0 * K_ + t];
    __syncthreads();

    // issue one chunk's async DMA: 128x32 floats for A (gathered) and B, b128 per lane
    auto issue = [&](int kb, int buf) {
        const int j     = kb >> 7;             // neighbor slot, constant in chunk
        const int cbase = kb & 127;
#pragma unroll
        for (int i = 0; i < 4; ++i) {
            int t  = tid + (i << 8);           // 0..1023 transfers of 16B
            int m  = t >> 3;                   // row 0..127 (8 xfers per row)
            int kk = (t & 7) << 2;             // float offset in row
            async_b128((uint32_t)(uintptr_t)&As[buf][m][kk],
                       F2_0 + (size_t)nb_s[m * K_ + j] * D_ + cbase + kk);
            async_b128((uint32_t)(uintptr_t)&Bs[buf][m][kk],
                       Wnb + (size_t)m * F_ + kb + kk);    // m doubles as d (both 128)
        }
    };

    const int lane = tid & 31;
    const int wv   = tid >> 5;                 // N-tile 0..7
    const int hf   = lane >> 4;                // half-wave 0/1
    const int mrow = lane & 15;
    const int ncol = lane & 15;
    const int dcol = wv * 16 + ncol;

    v8f acc[8];
#pragma unroll
    for (int mt = 0; mt < 8; ++mt) { v8f z = {}; acc[mt] = z; }

    issue(0, 0);
    for (int c = 0; c < NCHUNK; ++c) {
        wait_async0();                         // my async writes to LDS done
        __syncthreads();                       // => everyone's writes visible
        if (c + 1 < NCHUNK) issue((c + 1) * KC, (c + 1) & 1);   // overlap DMA w/ WMMA
        const int buf = c & 1;
        const float2* brow = reinterpret_cast<const float2*>(&Bs[buf][dcol][0]);
#pragma unroll
        for (int k = 0; k < KC; k += 4) {
            float2 bv = brow[(k >> 1) + hf];   // B 4x16 frag: K = k+2hf, k+2hf+1
            v2f bf; bf.x = bv.x; bf.y = bv.y;
#pragma unroll
            for (int mt = 0; mt < 8; ++mt) {   // 8 independent accumulation chains
                float2 av = *reinterpret_cast<const float2*>(
                    &As[buf][(mt << 4) + mrow][k + 2 * hf]);
                v2f af; af.x = av.x; af.y = av.y;
                acc[mt] = __builtin_amdgcn_wmma_f32_16x16x4_f32(
                    false, af, false, bf, (short)0, acc[mt], false, false);
            }
        }
    }

    // epilogue: + F3 + biases, leaky-relu, store into h[b, n+1, dcol]
    const float w30 = W3d[dcol * 3 + 0];
    const float w31 = W3d[dcol * 3 + 1];
    const float w32 = W3d[dcol * 3 + 2];
    const float bc  = b3d[dcol] + bnb[dcol];
    const int b = r0 >> 10;
#pragma unroll
    for (int mt = 0; mt < 8; ++mt) {
#pragma unroll
        for (int v = 0; v < 8; ++v) {
            int m = (mt << 4) + v + 8 * hf;    // C/D layout: VGPR v, half hf -> M
            int r = r0 + m;
            int n = r & (N_ - 1);
            float x0 = loc[r * 2 + 0];
            float x1 = loc[r * 2 + 1];
            float x2 = deadline[r];
            float f3 = fmaf(x0, w30, fmaf(x1, w31, fmaf(x2, w32, bc)));
            float val = acc[mt][v] + f3;
            val = val > 0.0f ? val : 0.01f * val;
            h[((size_t)b * (N_ + 1) + n + 1) * D_ + dcol] = val;
        }
    }
}

// ---------------- kernel 5: deterministic mean over axis 1 (1025 rows) ----------------
__global__ __launch_bounds__(128) void mean_kernel(const float* __restrict__ h,
                                                   float* __restrict__ meanout) {
    const int b = blockIdx.x;
    const int d = threadIdx.x;
    const float* p = h + (size_t)b * (N_ + 1) * D_ + d;
    float s = 0.0f;
    for (int i = 0; i <= N_; ++i) s += p[(size_t)i * D_];
    meanout[b * D_ + d] = s * (1.0f / (float)(N_ + 1));
}

extern "C" void kernel_launch(void* const* d_in, const int* in_sizes, int n_in,
                              void* d_out, int out_size, void* d_ws, size_t ws_size,
                              hipStream_t stream) {
    (void)in_sizes; (void)n_in; (void)out_size; (void)ws_size;
    const float* loc      = (const float*)d_in[0];
    const float* deadline = (const float*)d_in[1];
    const float* depot    = (const float*)d_in[2];
    const float* W3d      = (const float*)d_in[3];
    const float* b3d      = (const float*)d_in[4];
    const float* W2d      = (const float*)d_in[5];
    const float* b2d      = (const float*)d_in[6];
    const float* Wnb      = (const float*)d_in[7];
    const float* bnb      = (const float*)d_in[8];
    const float* Wdep     = (const float*)d_in[9];
    const float* bdep     = (const float*)d_in[10];

    float* h    = (float*)d_out;                       // (B, N+1, D)
    float* mn   = h + (size_t)B_ * (N_ + 1) * D_;      // (B, D)
    float* F2_0 = (float*)d_ws;                        // 1024*128 f32 = 512 KB
    int*   nbr  = (int*)((char*)d_ws + (size_t)N_ * D_ * sizeof(float)); // B*N*K ints

    f2_kernel <<<(N_ * D_ + 255) / 256, 256, 0, stream>>>(loc, W2d, b2d, F2_0);
    knn_kernel<<<B_ * 4, 256, 0, stream>>>(loc, nbr);
    dep_kernel<<<(B_ * D_ + 255) / 256, 256, 0, stream>>>(depot, Wdep, bdep, h);
    gemm_kernel<<<(B_ * N_) / MROWS, 256, 0, stream>>>(F2_0, nbr, Wnb, bnb,
                                                       loc, deadline, W3d, b3d, h);
    mean_kernel<<<B_, D_, 0, stream>>>(h, mn);
}